// MaskedFocalLoss_21371757265290
// MI455X (gfx1250) — compile-verified
//
#include <hip/hip_runtime.h>

// ---------------------------------------------------------------------------
// Masked focal loss, MI455X (gfx1250, wave32).
// Pass 1: streaming fused mask+focal partial reduction (bandwidth-bound).
// Pass 2: final 64-row reduction via V_WMMA_F32_16X16X4_F32.
// ---------------------------------------------------------------------------

typedef float v2f __attribute__((ext_vector_type(2)));
typedef float v8f __attribute__((ext_vector_type(8)));

namespace mfl {
constexpr int   Bn     = 64;
constexpr int   Sn     = 131072;
constexpr int   Cn     = 3;
constexpr int   WINDOW = 100;
constexpr int   CHUNK  = 4096;             // elements of S per block
constexpr int   BLOCKT = 256;              // 8 wave32 per block
constexpr int   CPR    = Sn / CHUNK;       // 32 chunks per row
constexpr int   PARTS  = Bn * CPR;         // 2048 partials
constexpr int   HALO_N = CHUNK + 2 * WINDOW;       // 4296 halo positions
constexpr int   NBITW  = (HALO_N + 31) / 32 + 1;   // 135 -> pad to 136
constexpr int   NITER  = (BLOCKT * 17 >= HALO_N) ? 17 : 18; // ballot iters
}

// ---------------------------------------------------------------------------
// Pass 1: per-(row, chunk) partials.
// ---------------------------------------------------------------------------
__global__ __launch_bounds__(mfl::BLOCKT)
void MaskedFocalLoss_partials(const float* __restrict__ inputs,
                              const int*   __restrict__ targets,
                              const float* __restrict__ alpha,
                              float* __restrict__ partMasked,
                              float* __restrict__ partFull,
                              int*   __restrict__ partCnt,
                              int*   __restrict__ partPos) {
    using namespace mfl;
    __shared__ unsigned s_bits[136];   // pos-flag bitmask over halo positions
    __shared__ float    s_ms[8], s_fs[8];
    __shared__ int      s_ct[8], s_pp[8];

    const int bid  = blockIdx.x;
    const int row  = bid / CPR;
    const int cidx = bid % CPR;
    const int tid  = threadIdx.x;
    const int lane = tid & 31;
    const int wid  = tid >> 5;
    const int chunkStart = cidx * CHUNK;

    const float a0 = alpha[0], a1 = alpha[1], a2 = alpha[2];
    const int* trow = targets + (size_t)row * Sn;

    // Build the positive-target bitmask for [chunkStart-100, chunkStart+4196)
    // with one wave32 ballot per 32 positions.
    for (int k = 0; k < NITER; ++k) {
        const int p = k * BLOCKT + tid;                 // halo-local position
        const int j = chunkStart - WINDOW + p;          // position in row
        int pred = 0;
        if (p < HALO_N && j >= 0 && j < Sn) pred = (trow[j] > 0) ? 1 : 0;
        const unsigned bal = (unsigned)__ballot(pred);  // wave32: low 32 bits
        if (lane == 0) s_bits[k * 8 + wid] = bal;
    }
    __syncthreads();

    float maskedSum = 0.0f;
    float fullSum   = 0.0f;
    int   cnt       = 0;
    int   anyPos    = 0;

    const float* in_base = inputs + (size_t)row * Sn * Cn;

    for (int it = 0; it < CHUNK / BLOCKT; ++it) {
        const int e = it * BLOCKT + tid;                // 0..4095
        const int i = chunkStart + e;

        // --- windowed-OR over bits [e, e+200] of the halo bitmask ---
        const int lo = e, hi = e + 2 * WINDOW;
        const int w0 = lo >> 5, w1 = hi >> 5;
        const unsigned mFirst = 0xFFFFFFFFu << (lo & 31);
        const unsigned mLast  = 0xFFFFFFFFu >> (31 - (hi & 31));
        unsigned acc;
        if (w0 == w1) {
            acc = s_bits[w0] & mFirst & mLast;
        } else {
            acc = s_bits[w0] & mFirst;
            for (int w = w0 + 1; w < w1; ++w) acc |= s_bits[w];
            acc |= s_bits[w1] & mLast;
        }
        const bool masked = (acc != 0u);

        // --- fused 3-class log-softmax + focal ---
        const float* p = in_base + (size_t)i * Cn;
        __builtin_prefetch(p + (size_t)BLOCKT * Cn, 0, 0); // global_prefetch_b8
        const float x0 = p[0], x1 = p[1], x2 = p[2];
        const int   t  = trow[i];

        const float mx = fmaxf(x0, fmaxf(x1, x2));
        const float e0 = expf(x0 - mx);
        const float e1 = expf(x1 - mx);
        const float e2 = expf(x2 - mx);
        const float sum = e0 + e1 + e2;
        const float xt = (t == 0) ? x0 : ((t == 1) ? x1 : x2);
        const float et = (t == 0) ? e0 : ((t == 1) ? e1 : e2);
        const float ce = (mx + logf(sum)) - xt;     // -log p_t
        const float pt = et / sum;                   // p_t
        const float om = 1.0f - pt;
        const float at = (t == 0) ? a0 : ((t == 1) ? a1 : a2);
        const float focal = at * om * om * ce;

        fullSum += focal;
        if (masked) { maskedSum += focal; ++cnt; }
        anyPos |= (t > 0);
    }

    // Deterministic wave32 tree reduction.
    for (int off = 16; off > 0; off >>= 1) {
        maskedSum += __shfl_xor(maskedSum, off, 32);
        fullSum   += __shfl_xor(fullSum,   off, 32);
        cnt       += __shfl_xor(cnt,       off, 32);
        anyPos    |= __shfl_xor(anyPos,    off, 32);
    }
    if (lane == 0) { s_ms[wid] = maskedSum; s_fs[wid] = fullSum;
                     s_ct[wid] = cnt;       s_pp[wid] = anyPos; }
    __syncthreads();

    if (tid == 0) {
        float ms = 0.0f, fs = 0.0f; int ct = 0, pp = 0;
        for (int w = 0; w < 8; ++w) {
            ms += s_ms[w]; fs += s_fs[w]; ct += s_ct[w]; pp |= s_pp[w];
        }
        partMasked[bid] = ms;
        partFull[bid]   = fs;
        partCnt[bid]    = ct;
        partPos[bid]    = pp;
    }
}

// ---------------------------------------------------------------------------
// Pass 2: one wave32. Per-row select (masked vs full), then a 64-way
// reduction through V_WMMA_F32_16X16X4_F32: the 64 per-row values map
// exactly onto the 16x4 A matrix; B = ones gives row sums in D; summing a
// lane's 8 D registers + (lane0 + lane16) yields the full sum.
// ---------------------------------------------------------------------------
__global__ __launch_bounds__(32)
void MaskedFocalLoss_finalize(const float* __restrict__ partMasked,
                              const float* __restrict__ partFull,
                              const int*   __restrict__ partCnt,
                              const int*   __restrict__ partPos,
                              float* __restrict__ out) {
    using namespace mfl;
    const int l = threadIdx.x;  // 0..31, EXEC all ones (WMMA requirement)

    // A[m][k] (16x4 f32): lanes 0-15 hold {K=0,K=1}, lanes 16-31 hold {K=2,K=3}.
    // Map row r = 4*m + k so the 64 rows fill A exactly.
    int r0, r1;
    if (l < 16) { r0 = 4 * l;        r1 = 4 * l + 1;        }
    else        { r0 = 4 * (l - 16) + 2; r1 = 4 * (l - 16) + 3; }

    float selSum[2], selCnt[2];
    const int rr[2] = { r0, r1 };
#pragma unroll
    for (int q = 0; q < 2; ++q) {
        const int r = rr[q];
        float ms = 0.0f, fs = 0.0f; int ct = 0, pp = 0;
        for (int k = 0; k < CPR; ++k) {
            const int idx = r * CPR + k;
            ms += partMasked[idx];
            fs += partFull[idx];
            ct += partCnt[idx];
            pp |= partPos[idx];
        }
        selSum[q] = pp ? ms : fs;
        selCnt[q] = pp ? (float)ct : (float)Sn;
    }

    const v2f Asum = { selSum[0], selSum[1] };
    const v2f Acnt = { selCnt[0], selCnt[1] };
    const v2f Bones = { 1.0f, 1.0f };
    const v8f Z = {};

    v8f Dsum = __builtin_amdgcn_wmma_f32_16x16x4_f32(
        false, Asum, false, Bones, (short)0, Z, false, false);
    v8f Dcnt = __builtin_amdgcn_wmma_f32_16x16x4_f32(
        false, Acnt, false, Bones, (short)0, Z, false, false);

    float s = 0.0f, c = 0.0f;
#pragma unroll
    for (int v = 0; v < 8; ++v) { s += Dsum[v]; c += Dcnt[v]; }
    // lane n<16 holds rows 0..7 (r=0..31); lane n>=16 holds rows 8..15 (r=32..63)
    const float s16 = __shfl(s, 16, 32);
    const float c16 = __shfl(c, 16, 32);
    if (l == 0) out[0] = (s + s16) / (c + c16);
}

// ---------------------------------------------------------------------------
extern "C" void kernel_launch(void* const* d_in, const int* in_sizes, int n_in,
                              void* d_out, int out_size, void* d_ws, size_t ws_size,
                              hipStream_t stream) {
    using namespace mfl;
    (void)in_sizes; (void)n_in; (void)out_size; (void)ws_size;

    const float* inputs  = (const float*)d_in[0];   // (B, S, 3) f32
    const int*   targets = (const int*)  d_in[1];   // (B, S)    i32
    const float* alpha   = (const float*)d_in[2];   // (3,)      f32
    float*       out     = (float*)d_out;

    float* partMasked = (float*)d_ws;               // PARTS f32
    float* partFull   = partMasked + PARTS;         // PARTS f32
    int*   partCnt    = (int*)(partFull + PARTS);   // PARTS i32
    int*   partPos    = partCnt + PARTS;            // PARTS i32  (32 KB total)

    MaskedFocalLoss_partials<<<PARTS, BLOCKT, 0, stream>>>(
        inputs, targets, alpha, partMasked, partFull, partCnt, partPos);

    MaskedFocalLoss_finalize<<<1, 32, 0, stream>>>(
        partMasked, partFull, partCnt, partPos, out);
}